// NFFTBase_77206332113308
// MI455X (gfx1250) — compile-verified
//
#include <hip/hip_runtime.h>
#include <math.h>

typedef __attribute__((ext_vector_type(2))) float v2f;
typedef __attribute__((ext_vector_type(8))) float v8f;

#define JW 6
#define IMN 320
#define GSZ 640
#define NC 12
#define MPTS 102400
#define PI_F 3.14159265358979f
#define KCHUNK 32
#define LDSP 17   // padded LDS row stride (floats) to kill bank conflicts

// BETA = pi * sqrt((J*(OSF-0.5)/OSF)^2 - 0.8) = pi*sqrt(4.5^2-0.8)
__device__ __forceinline__ float beta_const() {
    return PI_F * sqrtf(19.45f);
}

// Modified Bessel I0 via power series (converges for x <= ~14 in fp32)
__device__ __forceinline__ float i0f_dev(float x) {
    float t = 0.25f * x * x;
    float term = 1.0f, sum = 1.0f;
    #pragma unroll 1
    for (int k = 1; k <= 30; ++k) {
        term *= t / (float)(k * k);
        sum += term;
    }
    return sum;
}

// Kaiser-Bessel kernel, support |u| <= J/2 = 3
__device__ __forceinline__ float kbw(float u) {
    float m = fabsf(u) * (1.0f / 3.0f);   // |2u/J|
    if (m > 1.0f) return 0.0f;
    float s = 1.0f - m * m;
    s = s < 0.0f ? 0.0f : s;
    return i0f_dev(beta_const() * sqrtf(s)) * (1.0f / (float)JW);
}

// ---------------- apodization vector (length 320)
__global__ void apod_kernel(float* __restrict__ a) {
    int n = threadIdx.x;
    if (n >= IMN) return;
    float acc = 0.0f;
    for (int j = -JW; j <= JW; ++j) {
        float kv = kbw((float)j);
        float ang = 2.0f * PI_F * (float)j * (float)(n - IMN / 2) / (float)GSZ;
        acc += kv * cosf(ang);
    }
    a[n] = 1.0f / acc;
}

// ---------------- DFT matrix, planar: Fr/Fi[v*G+v'] = exp(-2*pi*i*v*v'/G)
__global__ void dft_init_kernel(float* __restrict__ Fr, float* __restrict__ Fi) {
    int i = blockIdx.x * blockDim.x + threadIdx.x;
    if (i >= GSZ * GSZ) return;
    int v = i / GSZ, vp = i - v * GSZ;
    int p = (int)(((long long)v * (long long)vp) % GSZ);  // exact phase index
    float ang = -2.0f * PI_F * (float)p / (float)GSZ;
    float s, c;
    __sincosf(ang, &s, &c);
    Fr[i] = c;
    Fi[i] = s;
}

// ---------------- coil image build + pad + ifftshift scatter (corners only), planar
__global__ void prep_kernel(const float* __restrict__ imr, const float* __restrict__ imi,
                            const float* __restrict__ smr, const float* __restrict__ smi,
                            const float* __restrict__ a,
                            float* __restrict__ Xr, float* __restrict__ Xi) {
    int i = blockIdx.x * blockDim.x + threadIdx.x;
    if (i >= NC * IMN * IMN) return;
    int c = i / (IMN * IMN);
    int rem = i - c * IMN * IMN;
    int r = rem / IMN, s = rem - r * IMN;
    float sc = a[r] * a[s];
    float xr = imr[rem] * sc, xi = imi[rem] * sc;
    float sr = smr[i], si = smi[i];
    float vr = xr * sr - xi * si;
    float vi = xr * si + xi * sr;
    int u = (r < 160) ? r + 480 : r - 160;
    int v = (s < 160) ? s + 480 : s - 160;
    size_t o = ((size_t)c * GSZ + u) * GSZ + v;
    Xr[o] = vr;
    Xi[o] = vi;
}

// ---------------- complex GEMM vs DFT matrix via v_wmma_f32_16x16x4_f32.
// Planar operands. 4 waves/WG share an LDS-staged 16x32 A panel (transposed,
// padded) and compute 4 adjacent 16x16 N-tiles. K runs over the 320 nonzero
// indices kk -> k = kk<160 ? kk : kk+320.
__global__ void __launch_bounds__(128) cgemm_dft_kernel(
    const float* __restrict__ Ar_g, const float* __restrict__ Ai_g, int aCoilStride, int aRowRemap,
    const float* __restrict__ Br_g, const float* __restrict__ Bi_g, int bCoilStride,
    float* __restrict__ Cr_g, float* __restrict__ Ci_g, float2* __restrict__ Cz_g,
    int cCoilStride, float outScale) {
    __shared__ float sAr[KCHUNK * LDSP];
    __shared__ float sAi[KCHUNK * LDSP];

    const int tid  = threadIdx.x;
    const int lane = tid & 31;
    const int wave = tid >> 5;
    const int lo = lane & 15;
    const int hi = lane >> 4;
    const int coil = blockIdx.z;

    const float* Ar = Ar_g + (size_t)coil * aCoilStride;
    const float* Ai = Ai_g + (size_t)coil * aCoilStride;
    const float* Br = Br_g + (size_t)coil * bCoilStride;
    const float* Bi = Bi_g + (size_t)coil * bCoilStride;

    const int tM = blockIdx.y;
    const int u0 = aRowRemap ? (tM < 10 ? tM * 16 : 480 + (tM - 10) * 16) : tM * 16;
    const int n0 = (blockIdx.x * 4 + wave) * 16;
    const int n  = n0 + lo;

    v8f cr = {};
    v8f ci = {};

    for (int chunk = 0; chunk < 10; ++chunk) {
        const int kkb = chunk * KCHUNK;
        const int kb  = (kkb < 160) ? kkb : kkb + 320;  // remapped, contiguous block of 32

        // cooperative fill: 512 elements/plane, coalesced along k, transposed into LDS
        #pragma unroll
        for (int p = 0; p < 4; ++p) {
            int idx = p * 128 + tid;
            int row = idx >> 5;        // 0..15
            int kl  = idx & 31;        // 0..31
            size_t go = (size_t)(u0 + row) * GSZ + (kb + kl);
            sAr[kl * LDSP + row] = Ar[go];
            sAi[kl * LDSP + row] = Ai[go];
        }
        __syncthreads();

        #pragma unroll 2
        for (int s = 0; s < 8; ++s) {
            const int kl = 4 * s;
            const int k  = kb + kl;
            // A fragment 16x4 from LDS (bank-conflict-free)
            v2f ar = { sAr[(kl + 2 * hi) * LDSP + lo], sAr[(kl + 2 * hi + 1) * LDSP + lo] };
            v2f ai = { sAi[(kl + 2 * hi) * LDSP + lo], sAi[(kl + 2 * hi + 1) * LDSP + lo] };
            v2f nai = -ai;
            // B fragment 4x16, planar, coalesced b32 loads
            v2f br = { Br[(size_t)(k + 2 * hi) * GSZ + n], Br[(size_t)(k + 2 * hi + 1) * GSZ + n] };
            v2f bi = { Bi[(size_t)(k + 2 * hi) * GSZ + n], Bi[(size_t)(k + 2 * hi + 1) * GSZ + n] };
            // Cr += Ar*Br - Ai*Bi ; Ci += Ar*Bi + Ai*Br
            cr = __builtin_amdgcn_wmma_f32_16x16x4_f32(false, ar, false, br, (short)0, cr, false, false);
            cr = __builtin_amdgcn_wmma_f32_16x16x4_f32(false, nai, false, bi, (short)0, cr, false, false);
            ci = __builtin_amdgcn_wmma_f32_16x16x4_f32(false, ar, false, bi, (short)0, ci, false, false);
            ci = __builtin_amdgcn_wmma_f32_16x16x4_f32(false, ai, false, br, (short)0, ci, false, false);
        }
        __syncthreads();
    }

    if (Cz_g) {  // interleaved complex output (final k-space grid)
        float2* Cz = Cz_g + (size_t)coil * cCoilStride;
        #pragma unroll
        for (int g = 0; g < 8; ++g) {
            int mrow = u0 + g + 8 * hi;
            Cz[(size_t)mrow * GSZ + n] = make_float2(cr[g] * outScale, ci[g] * outScale);
        }
    } else {     // planar output (intermediate Y)
        float* Cr = Cr_g + (size_t)coil * cCoilStride;
        float* Ci = Ci_g + (size_t)coil * cCoilStride;
        #pragma unroll
        for (int g = 0; g < 8; ++g) {
            int mrow = u0 + g + 8 * hi;
            size_t o = (size_t)mrow * GSZ + n;
            Cr[o] = cr[g] * outScale;
            Ci[o] = ci[g] * outScale;
        }
    }
}

// ---------------- KB gridding interpolation: one thread per trajectory point
__global__ void interp_kernel(const float* __restrict__ ktraj,
                              const float2* __restrict__ Zg,
                              float* __restrict__ out) {
    int m = blockIdx.x * blockDim.x + threadIdx.x;
    if (m >= MPTS) return;
    float om0 = ktraj[m];
    float om1 = ktraj[MPTS + m];

    int ix[JW], iy[JW];
    float wx[JW], wy[JW];
    {
        float t = om0 * ((float)GSZ / (2.0f * PI_F));
        t = fmodf(t, (float)GSZ); if (t < 0.0f) t += (float)GSZ;
        float base = floorf(t);
        #pragma unroll
        for (int j = 0; j < JW; ++j) {
            float kf = base + (float)(j - 2);
            wx[j] = kbw(t - kf);
            int ki = (int)kf; ki %= GSZ; if (ki < 0) ki += GSZ;
            ix[j] = ki;
        }
    }
    {
        float t = om1 * ((float)GSZ / (2.0f * PI_F));
        t = fmodf(t, (float)GSZ); if (t < 0.0f) t += (float)GSZ;
        float base = floorf(t);
        #pragma unroll
        for (int j = 0; j < JW; ++j) {
            float kf = base + (float)(j - 2);
            wy[j] = kbw(t - kf);
            int ki = (int)kf; ki %= GSZ; if (ki < 0) ki += GSZ;
            iy[j] = ki;
        }
    }
    for (int c = 0; c < NC; ++c) {
        const float2* gc = Zg + (size_t)c * GSZ * GSZ;
        float sr = 0.0f, si = 0.0f;
        #pragma unroll
        for (int i = 0; i < JW; ++i) {
            const float2* row = gc + (size_t)ix[i] * GSZ;
            float wxi = wx[i];
            #pragma unroll
            for (int j = 0; j < JW; ++j) {
                float w = wxi * wy[j];
                float2 v = row[iy[j]];
                sr += w * v.x;
                si += w * v.y;
            }
        }
        size_t o = ((size_t)c * MPTS + m) * 2;
        out[o] = sr;
        out[o + 1] = si;
    }
}

extern "C" void kernel_launch(void* const* d_in, const int* in_sizes, int n_in,
                              void* d_out, int out_size, void* d_ws, size_t ws_size,
                              hipStream_t stream) {
    const float* image_r = (const float*)d_in[0];   // [1,320,320]
    const float* image_i = (const float*)d_in[1];   // [1,320,320]
    const float* smaps_r = (const float*)d_in[2];   // [1,12,320,320]
    const float* smaps_i = (const float*)d_in[3];   // [1,12,320,320]
    const float* ktraj   = (const float*)d_in[4];   // [1,2,102400]
    float* out = (float*)d_out;                     // [1,12,102400,2]

    const size_t fplane = (size_t)GSZ * GSZ * 4;          // 1.64 MB
    const size_t cplane = (size_t)NC * GSZ * GSZ * 4;     // 19.66 MB

    char* ws = (char*)d_ws;
    const size_t offFr = 4096;
    const size_t offFi = offFr + fplane;
    const size_t offXr = offFi + fplane;
    const size_t offXi = offXr + cplane;
    const size_t offYr = offXi + cplane;
    const size_t offYi = offYr + cplane;                  // total ~82 MB

    float*  apod = (float*)(ws);
    float*  Fr   = (float*)(ws + offFr);
    float*  Fi   = (float*)(ws + offFi);
    float*  Xr   = (float*)(ws + offXr);
    float*  Xi   = (float*)(ws + offXi);
    float*  Yr   = (float*)(ws + offYr);
    float*  Yi   = (float*)(ws + offYi);
    float2* Z    = (float2*)(ws + offXr);  // final grid aliases X planes (X dead by stage 2)

    // 1) apodization correction vector
    apod_kernel<<<1, IMN, 0, stream>>>(apod);

    // 2) DFT matrix (planar)
    dft_init_kernel<<<(GSZ * GSZ + 255) / 256, 256, 0, stream>>>(Fr, Fi);

    // 3) coil images -> padded/ifftshifted grid corners (planar)
    prep_kernel<<<(NC * IMN * IMN + 255) / 256, 256, 0, stream>>>(
        image_r, image_i, smaps_r, smaps_i, apod, Xr, Xi);

    // 4) stage 1: Y = X * F  (row DFT; only 320 nonzero rows per coil), planar out
    {
        dim3 grid(GSZ / 64, 20, NC);   // 10 blocks x 4 waves = 40 n-tiles; 20 row-tiles
        cgemm_dft_kernel<<<grid, 128, 0, stream>>>(
            Xr, Xi, GSZ * GSZ, /*aRowRemap=*/1,
            Fr, Fi, 0,
            Yr, Yi, nullptr, GSZ * GSZ, 1.0f);
    }

    // 5) stage 2: Z = F * Y  (col DFT), 1/640 normalization, interleaved out
    {
        dim3 grid(GSZ / 64, GSZ / 16, NC);
        cgemm_dft_kernel<<<grid, 128, 0, stream>>>(
            Fr, Fi, 0, /*aRowRemap=*/0,
            Yr, Yi, GSZ * GSZ,
            nullptr, nullptr, Z, GSZ * GSZ, 1.0f / (float)GSZ);
    }

    // 6) Kaiser-Bessel interpolation at trajectory points
    interp_kernel<<<(MPTS + 255) / 256, 256, 0, stream>>>(ktraj, Z, out);
}